// SpectralRegressor_27487790695174
// MI455X (gfx1250) — compile-verified
//
#include <hip/hip_runtime.h>

#define BATCH   16
#define NGRID   8192
#define HID     128
#define MODES   32
#define NLAYERS 4
#define DFF     256
#define KSPLIT  8
#define KCHUNK  256
#define LDA     258   // padded LDS row stride (floats) to dodge bank conflicts

typedef __attribute__((ext_vector_type(2))) float v2f;
typedef __attribute__((ext_vector_type(8))) float v8f;

__device__ __forceinline__ v8f wmma4(v2f a, v2f b, v8f c) {
  return __builtin_amdgcn_wmma_f32_16x16x4_f32(false, a, false, b, (short)0, c,
                                               false, false);
}

__device__ __forceinline__ float silu_f(float v) {
  return v * (1.0f / (1.0f + __expf(-v)));
}

__device__ __forceinline__ v8f zero8() {
  v8f z = {0.f, 0.f, 0.f, 0.f, 0.f, 0.f, 0.f, 0.f};
  return z;
}

// CDNA5 async copy global->LDS (ASYNCcnt-tracked), see CDNA5 ISA
__device__ __forceinline__ void async_copy_b128(unsigned lds_byte_addr,
                                                const float* gaddr) {
  asm volatile("global_load_async_to_lds_b128 %0, %1, off"
               :: "v"(lds_byte_addr), "v"(gaddr) : "memory");
}
__device__ __forceinline__ void async_wait0() {
  asm volatile("s_wait_asynccnt 0x0" ::: "memory");
}

// ---------------------------------------------------------------------------
// Truncated DFT bases.
// fwdBas[2m][n] = cos(2pi m n/N), fwdBas[2m+1][n] = -sin(...)   ([64][N])
// invBas[n][2m] = cos(2pi m n/N), invBas[n][2m+1] = -sin(...)   ([N][64])
// ---------------------------------------------------------------------------
__global__ void basis_kernel(float* __restrict__ fwdBas, float* __restrict__ invBas) {
  int idx = blockIdx.x * blockDim.x + threadIdx.x;
  if (idx >= NGRID * MODES) return;
  int n = idx >> 5;
  int m = idx & (MODES - 1);
  int t = (n * m) & (NGRID - 1);
  float theta = (float)t * (6.283185307179586f / (float)NGRID);
  float s, c;
  __sincosf(theta, &s, &c);
  fwdBas[(2 * m) * NGRID + n]     = c;
  fwdBas[(2 * m + 1) * NGRID + n] = -s;
  invBas[n * (2 * MODES) + 2 * m]     = c;
  invBas[n * (2 * MODES) + 2 * m + 1] = -s;
}

// ---------------------------------------------------------------------------
// Forward truncated DFT, K split 8 ways for occupancy:
//   xftPart[p][b][c][i] = sum_{n in part p} fwdBas[c][n] * x[b][n][i]
// grid(4, B, 8), block 256. Basis chunk staged in LDS via async-to-LDS.
// ---------------------------------------------------------------------------
__global__ void dft_kernel(const float* __restrict__ x, const float* __restrict__ fwdBas,
                           float* __restrict__ xftPart) {
  __shared__ float ldsA[16 * LDA];
  const int b    = blockIdx.y;
  const int c0   = blockIdx.x * 16;
  const int part = blockIdx.z;
  const int tid  = threadIdx.x;
  const int wave = tid >> 5;
  const int lane = tid & 31;
  const int half = lane >> 4;
  const int ln16 = lane & 15;
  const int i0   = wave * 16;

  const float* xb  = x + (size_t)b * NGRID * HID;
  const int kbeg   = part * (NGRID / KSPLIT);
  const int crow   = tid >> 4;           // staging: row 0..15
  const int cseg   = (tid & 15) * 16;    // staging: 16-float segment
  const unsigned ldsBase = (unsigned)(size_t)(&ldsA[0]);

  v8f acc = zero8();
  for (int kc = 0; kc < NGRID / KSPLIT; kc += KCHUNK) {
    const int kbase = kbeg + kc;
    {
      const float* gsrc = fwdBas + (size_t)(c0 + crow) * NGRID + kbase + cseg;
      const unsigned ldst = ldsBase + (unsigned)((crow * LDA + cseg) * 4);
      #pragma unroll
      for (int q = 0; q < 4; ++q)
        async_copy_b128(ldst + q * 16, gsrc + q * 4);
    }
    async_wait0();
    __syncthreads();
    #pragma unroll 4
    for (int k0 = 0; k0 < KCHUNK; k0 += 4) {
      const int kk = k0 + half * 2;
      v2f a = *(const v2f*)&ldsA[ln16 * LDA + kk];
      const size_t gk = (size_t)(kbase + kk) * HID + i0 + ln16;
      v2f bb; bb.x = xb[gk];
              bb.y = xb[gk + HID];
      acc = wmma4(a, bb, acc);
    }
    __syncthreads();
  }
  float* ob = xftPart + ((size_t)part * BATCH + b) * 64 * HID;
  #pragma unroll
  for (int j = 0; j < 8; ++j)
    ob[(c0 + half * 8 + j) * HID + i0 + ln16] = acc[j];
}

// Deterministic fixed-order reduction of the 8 K-partials.
__global__ void reduce_kernel(const float* __restrict__ xftPart, float* __restrict__ xft) {
  const int idx = blockIdx.x * blockDim.x + threadIdx.x;  // < B*64*HID
  float s = 0.f;
  #pragma unroll
  for (int p = 0; p < KSPLIT; ++p)
    s += xftPart[(size_t)p * BATCH * 64 * HID + idx];
  xft[idx] = s;
}

// ---------------------------------------------------------------------------
// Complex mode mixing (67 MFLOP, VALU) + fold ortho/Hermitian scaling:
//   oft[b][2m][o] + i*oft[b][2m+1][o] = s_m * sum_i xft_c[b][i][m] * w_c[i][o][m]
//   s_m = (m==0 ? 1 : 2)/N
// ---------------------------------------------------------------------------
__global__ void modemix_kernel(const float* __restrict__ xft, const float* __restrict__ fw,
                               float* __restrict__ oft) {
  const int m = blockIdx.x;
  const int b = blockIdx.y;
  const int o = threadIdx.x;   // 128
  __shared__ float xr[HID], xi[HID];
  const float* xb = xft + (size_t)b * 64 * HID;
  xr[o] = xb[(2 * m) * HID + o];
  xi[o] = xb[(2 * m + 1) * HID + o];
  __syncthreads();
  float re = 0.f, im = 0.f;
  for (int i = 0; i < HID; ++i) {
    const size_t wo = (size_t)(i * HID + o) * (2 * MODES) + 2 * m;
    const float wr = fw[wo];
    const float wi = fw[wo + 1];
    re = fmaf(xr[i], wr, re);
    re = fmaf(-xi[i], wi, re);
    im = fmaf(xr[i], wi, im);
    im = fmaf(xi[i], wr, im);
  }
  const float s = (m == 0 ? 1.0f : 2.0f) / (float)NGRID;
  float* ob = oft + (size_t)b * 64 * HID;
  ob[(2 * m) * HID + o]     = s * re;
  ob[(2 * m + 1) * HID + o] = s * im;
}

// ---------------------------------------------------------------------------
// Fused layer: y = silu( x@lw + lb + invDFT(oft) ), one GEMM with K=192:
//   A_cat(n,k) = k<128 ? x[b][n][k] : invBas[n][k-128]
//   B_cat(k,o) = k<128 ? lw[k][o]   : oft[b][k-128][o]
// B_cat staged in LDS in three contiguous 32KB chunks via async-to-LDS;
// each of the 8 waves then reads B fragments from LDS (8x less VMEM).
// In-place safe: block reads only its own rows, writes after all reads.
// ---------------------------------------------------------------------------
__global__ void layer_kernel(const float* __restrict__ xin, float* __restrict__ xout,
                             const float* __restrict__ lw, const float* __restrict__ lb,
                             const float* __restrict__ oft, const float* __restrict__ invBas) {
  __shared__ float ldsB[64 * HID];   // 32KB chunk of B_cat
  const int b    = blockIdx.y;
  const int tid  = threadIdx.x;
  const int wave = tid >> 5;
  const int lane = tid & 31;
  const int half = lane >> 4;
  const int ln16 = lane & 15;
  const int n0   = blockIdx.x * 128 + wave * 16;

  const float* xrow = xin + ((size_t)b * NGRID + n0) * HID;
  const float* ob   = oft + (size_t)b * 64 * HID;
  const float* irow = invBas + (size_t)(n0 + ln16) * 64;
  const unsigned ldsBase = (unsigned)(size_t)(&ldsB[0]);

  v8f acc[8];
  #pragma unroll
  for (int f = 0; f < 8; ++f) acc[f] = zero8();

  for (int chunk = 0; chunk < 3; ++chunk) {
    const float* gsrc = (chunk == 0) ? lw : (chunk == 1) ? (lw + 64 * HID) : ob;
    {
      const float* g = gsrc + tid * 4;
      const unsigned l0 = ldsBase + (unsigned)(tid * 16);
      #pragma unroll
      for (int q = 0; q < 8; ++q)                 // 8 x 4KB = 32KB per block
        async_copy_b128(l0 + q * 4096, g + q * 1024);
    }
    async_wait0();
    __syncthreads();

    const float* aPtr = (chunk < 2) ? (xrow + ln16 * HID + chunk * 64) : irow;
    #pragma unroll 2
    for (int k0 = 0; k0 < 64; k0 += 4) {
      const int kk = k0 + half * 2;
      v2f a; a.x = aPtr[kk];
             a.y = aPtr[kk + 1];
      #pragma unroll
      for (int f = 0; f < 8; ++f) {
        v2f bb; bb.x = ldsB[kk * HID + f * 16 + ln16];
                bb.y = ldsB[(kk + 1) * HID + f * 16 + ln16];
        acc[f] = wmma4(a, bb, acc[f]);
      }
    }
    __syncthreads();
  }

  float* yrow = xout + ((size_t)b * NGRID + n0) * HID;
  #pragma unroll
  for (int f = 0; f < 8; ++f) {
    const int col = f * 16 + ln16;
    const float bias = lb[col];
    #pragma unroll
    for (int j = 0; j < 8; ++j)
      yrow[(half * 8 + j) * HID + col] = silu_f(acc[f][j] + bias);
  }
}

// ---------------------------------------------------------------------------
// Regressor: out = silu(x@W1 + b1) @ W2 + b2, fused. A strip cached in regs,
// per f-tile WMMA chain (K=128), SiLU, scale by W2, 16-lane shuffle reduce.
// ---------------------------------------------------------------------------
__global__ void regressor_kernel(const float* __restrict__ xin, const float* __restrict__ w1,
                                 const float* __restrict__ b1, const float* __restrict__ w2,
                                 const float* __restrict__ b2, float* __restrict__ out) {
  const int b    = blockIdx.y;
  const int wave = threadIdx.x >> 5;
  const int lane = threadIdx.x & 31;
  const int half = lane >> 4;
  const int ln16 = lane & 15;
  const int n0   = blockIdx.x * 128 + wave * 16;
  const float* xrow = xin + ((size_t)b * NGRID + n0) * HID;

  v2f aReg[32];
  #pragma unroll
  for (int ks = 0; ks < 32; ++ks) {
    const int kk = ks * 4 + half * 2;
    aReg[ks].x = xrow[ln16 * HID + kk];
    aReg[ks].y = xrow[ln16 * HID + kk + 1];
  }
  float accOut[8] = {0.f, 0.f, 0.f, 0.f, 0.f, 0.f, 0.f, 0.f};

  for (int ft = 0; ft < DFF / 16; ++ft) {
    v8f c = zero8();
    #pragma unroll 4
    for (int ks = 0; ks < 32; ++ks) {
      const int kk = ks * 4 + half * 2;
      v2f bb; bb.x = w1[kk * DFF + ft * 16 + ln16];
              bb.y = w1[(kk + 1) * DFF + ft * 16 + ln16];
      c = wmma4(aReg[ks], bb, c);
    }
    const float b1v = b1[ft * 16 + ln16];
    const float w2v = w2[ft * 16 + ln16];
    #pragma unroll
    for (int j = 0; j < 8; ++j) {
      const float h = silu_f(c[j] + b1v);
      accOut[j] = fmaf(h, w2v, accOut[j]);
    }
  }
  #pragma unroll
  for (int j = 0; j < 8; ++j) {
    float v = accOut[j];
    v += __shfl_xor(v, 1);
    v += __shfl_xor(v, 2);
    v += __shfl_xor(v, 4);
    v += __shfl_xor(v, 8);
    if (ln16 == 0)
      out[(size_t)b * NGRID + n0 + half * 8 + j] = v + b2[0];
  }
}

// ---------------------------------------------------------------------------
extern "C" void kernel_launch(void* const* d_in, const int* in_sizes, int n_in,
                              void* d_out, int out_size, void* d_ws, size_t ws_size,
                              hipStream_t stream) {
  const float* x      = (const float*)d_in[0];
  const float* lin_w  = (const float*)d_in[1];
  const float* lin_b  = (const float*)d_in[2];
  const float* four_w = (const float*)d_in[3];
  const float* reg_w1 = (const float*)d_in[4];
  const float* reg_b1 = (const float*)d_in[5];
  const float* reg_w2 = (const float*)d_in[6];
  const float* reg_b2 = (const float*)d_in[7];
  float* out = (float*)d_out;

  // workspace (floats): xbuf | fwdBas | invBas | xft | oft | xftPart (~77MB)
  float* xbuf    = (float*)d_ws;
  float* fwdBas  = xbuf + (size_t)BATCH * NGRID * HID;
  float* invBas  = fwdBas + (size_t)64 * NGRID;
  float* xft     = invBas + (size_t)NGRID * 64;
  float* oft     = xft + (size_t)BATCH * 64 * HID;
  float* xftPart = oft + (size_t)BATCH * 64 * HID;

  basis_kernel<<<(NGRID * MODES + 255) / 256, 256, 0, stream>>>(fwdBas, invBas);

  for (int l = 0; l < NLAYERS; ++l) {
    const float* src = (l == 0) ? x : xbuf;
    dft_kernel<<<dim3(4, BATCH, KSPLIT), 256, 0, stream>>>(src, fwdBas, xftPart);
    reduce_kernel<<<(BATCH * 64 * HID) / 256, 256, 0, stream>>>(xftPart, xft);
    modemix_kernel<<<dim3(MODES, BATCH), HID, 0, stream>>>(
        xft, four_w + (size_t)l * HID * HID * MODES * 2, oft);
    layer_kernel<<<dim3(NGRID / 128, BATCH), 256, 0, stream>>>(
        src, xbuf, lin_w + (size_t)l * HID * HID, lin_b + (size_t)l * HID, oft, invBas);
  }
  regressor_kernel<<<dim3(NGRID / 128, BATCH), 256, 0, stream>>>(
      xbuf, reg_w1, reg_b1, reg_w2, reg_b2, out);
}